// StiffnessMatrix_12799002542408
// MI455X (gfx1250) — compile-verified
//
#include <hip/hip_runtime.h>

typedef __attribute__((ext_vector_type(2))) float v2f;
typedef __attribute__((ext_vector_type(4))) float v4f;
typedef __attribute__((ext_vector_type(8))) float v8f;

// ---------------------------------------------------------------------------
// Constant coefficient table: Kmat[e, 6r+c] = sum_k x[e,k] * B[k, 6r+c]
// features k: 0:krot*s2 1:krot*sc 2:krot*c2 3:krot*L*s 4:krot*L*c 5:krot*L^2
//             6:klin*c2 7:klin*sc 8:klin*s2
// ---------------------------------------------------------------------------
struct Coef { unsigned char k, r, c; signed char v; };

__constant__ Coef kCoefs[52] = {
    // f0 = krot*s2
    {0,0,0,12},{0,0,3,-12},{0,3,0,-12},{0,3,3,12},
    // f1 = krot*sc
    {1,0,1,12},{1,0,4,-12},{1,1,0,12},{1,1,3,-12},
    {1,3,1,-12},{1,3,4,12},{1,4,0,-12},{1,4,3,12},
    // f2 = krot*c2
    {2,1,1,12},{2,1,4,-12},{2,4,1,-12},{2,4,4,12},
    // f3 = krot*L*sin (Ls = 6*L*sin -> coeff +-6)
    {3,0,2,-6},{3,0,5,-6},{3,2,0,-6},{3,2,3,6},
    {3,3,2,6},{3,3,5,6},{3,5,0,-6},{3,5,3,6},
    // f4 = krot*L*cos
    {4,1,2,-6},{4,1,5,-6},{4,2,1,-6},{4,2,4,6},
    {4,4,2,6},{4,4,5,6},{4,5,1,-6},{4,5,4,6},
    // f5 = krot*L^2 (L4 = 4L^2, L2 = 2L^2)
    {5,2,2,4},{5,2,5,2},{5,5,2,2},{5,5,5,4},
    // f6 = klin*c2
    {6,0,0,1},{6,0,3,-1},{6,3,0,-1},{6,3,3,1},
    // f7 = klin*sc
    {7,0,1,-1},{7,0,4,1},{7,1,0,-1},{7,1,3,1},
    {7,3,1,1},{7,3,4,-1},{7,4,0,1},{7,4,3,-1},
    // f8 = klin*s2
    {8,1,1,1},{8,1,4,-1},{8,4,1,-1},{8,4,4,1},
};

__device__ __forceinline__ v2f mk2(float a, float b) { v2f r; r.x = a; r.y = b; return r; }

// ---------------------------------------------------------------------------
// Kernel 1: zero the 324MB output with wide nontemporal stores (BW-bound part)
// ---------------------------------------------------------------------------
__global__ void zero_out_kernel(v4f* __restrict__ out4, long n4,
                                float* __restrict__ out, long total) {
    long tid    = (long)blockIdx.x * blockDim.x + threadIdx.x;
    long stride = (long)gridDim.x * blockDim.x;
    v4f z = {0.f, 0.f, 0.f, 0.f};
    for (long i = tid; i < n4; i += stride)
        __builtin_nontemporal_store(z, &out4[i]);
    for (long i = 4 * n4 + tid; i < total; i += stride)
        out[i] = 0.f;
}

// ---------------------------------------------------------------------------
// Kernel 2: per-edge 6x6 blocks via V_WMMA_F32_16X16X4_F32, then atomic scatter
//   wave w handles 16 edges: A = features [16 x 12(K)], B = const [12 x 48]
//   D tile n: lane l holds cols 16n + l%16 of edges M = 8*(l/16) + 0..7
// ---------------------------------------------------------------------------
__global__ void assemble_kernel(const float* __restrict__ coord,
                                const float* __restrict__ delta,
                                const float* __restrict__ E,
                                const float* __restrict__ A,
                                const float* __restrict__ I,
                                const int*   __restrict__ src,
                                const int*   __restrict__ dst,
                                float* __restrict__ out,
                                int n_edges, int dim) {
    // B staged in LDS, packed per (tile, kstep, lane) as the WMMA B operand:
    // sB[tile][t][lane] = { B[4t + 2*(lane/16) + 0][16*tile + lane%16],
    //                       B[4t + 2*(lane/16) + 1][16*tile + lane%16] }
    __shared__ v2f sB[3][3][32];
    float* sf = (float*)sB;                     // 576 floats
    for (int i = threadIdx.x; i < 576; i += blockDim.x) sf[i] = 0.f;
    __syncthreads();
    if (threadIdx.x < 52) {
        Coef cf  = kCoefs[threadIdx.x];
        int col  = cf.r * 6 + cf.c;
        int t    = cf.k >> 2;
        int kr   = cf.k & 3;
        int lane = ((kr >> 1) << 4) + (col & 15);
        int tile = col >> 4;
        int sub  = kr & 1;
        sf[(((tile * 3 + t) << 5) + lane) * 2 + sub] = (float)cf.v;
    }
    __syncthreads();

    const int wave = threadIdx.x >> 5;
    const int lane = threadIdx.x & 31;
    const int e0   = (blockIdx.x * 8 + wave) * 16;   // 16 edges per wave
    if (e0 >= n_edges) return;                        // whole-wave uniform exit

    const int hi   = lane >> 4;    // 0: K pair {4t,4t+1}; 1: K pair {4t+2,4t+3}
    const int lsub = lane & 15;

    // ---- per-edge features (lane l and l+16 compute the same edge) ----
    const int e  = e0 + lsub;
    const int s  = src[e];
    const int d  = dst[e];
    const float csx = coord[2 * s]     + delta[2 * s];
    const float csy = coord[2 * s + 1] + delta[2 * s + 1];
    const float cdx = coord[2 * d]     + delta[2 * d];
    const float cdy = coord[2 * d + 1] + delta[2 * d + 1];
    const float dx = csx - cdx, dy = csy - cdy;
    const float L    = sqrtf(dx * dx + dy * dy);
    const float invL = 1.0f / L;
    const float co   = dx * invL;
    const float si   = -dy * invL;
    const float krot = E[e] * I[e] * invL * invL * invL;
    const float klin = E[e] * A[e] * invL;
    const float x0 = krot * si * si, x1 = krot * si * co, x2 = krot * co * co;
    const float x3 = krot * L * si,  x4 = krot * L * co,  x5 = krot * L * L;
    const float x6 = klin * co * co, x7 = klin * si * co, x8 = klin * si * si;

    // A operands for the three K-steps (ISA A layout for 32-bit 16x4 tiles)
    const v2f a0 = hi ? mk2(x2, x3)  : mk2(x0, x1);   // K = 0..3
    const v2f a1 = hi ? mk2(x6, x7)  : mk2(x4, x5);   // K = 4..7
    const v2f a2 = hi ? mk2(0.f,0.f) : mk2(x8, 0.f);  // K = 8..11 (9..11 zero)

    // src/dst of the 8 edges this lane scatters (M = 8*hi + m)
    const int eb = e0 + 8 * hi;
    int s8[8], d8[8];
#pragma unroll
    for (int m = 0; m < 8; ++m) { s8[m] = src[eb + m]; d8[m] = dst[eb + m]; }

#pragma unroll
    for (int tile = 0; tile < 3; ++tile) {
        v8f acc = {};
        acc = __builtin_amdgcn_wmma_f32_16x16x4_f32(false, a0, false, sB[tile][0][lane],
                                                    (short)0, acc, false, false);
        acc = __builtin_amdgcn_wmma_f32_16x16x4_f32(false, a1, false, sB[tile][1][lane],
                                                    (short)0, acc, false, false);
        acc = __builtin_amdgcn_wmma_f32_16x16x4_f32(false, a2, false, sB[tile][2][lane],
                                                    (short)0, acc, false, false);
        const int col = tile * 16 + lsub;         // flattened (r,c) = 6r + c
        if (col < 36) {
            const int rr = col / 6, cc = col % 6;
#pragma unroll
            for (int m = 0; m < 8; ++m) {
                const int gro = (rr < 3) ? (3 * s8[m] + rr) : (3 * d8[m] + rr - 3);
                const int gco = (cc < 3) ? (3 * s8[m] + cc) : (3 * d8[m] + cc - 3);
                atomicAdd(out + (size_t)gro * (size_t)dim + (size_t)gco, acc[m]);
            }
        }
    }
}

// ---------------------------------------------------------------------------
extern "C" void kernel_launch(void* const* d_in, const int* in_sizes, int n_in,
                              void* d_out, int out_size, void* d_ws, size_t ws_size,
                              hipStream_t stream) {
    const float* coord = (const float*)d_in[0];
    const float* delta = (const float*)d_in[1];
    const float* E     = (const float*)d_in[2];
    const float* A     = (const float*)d_in[3];
    const float* I     = (const float*)d_in[4];
    const int*   src   = (const int*)d_in[5];
    const int*   dst   = (const int*)d_in[6];
    float* out = (float*)d_out;

    const int n_nodes = in_sizes[0] / 2;
    const int n_edges = in_sizes[2];
    const int dim     = 3 * n_nodes;

    // 1) zero the output (dominant cost: ~324MB of stores, HBM-bound)
    const long total = (long)out_size;
    const long n4    = total >> 2;
    zero_out_kernel<<<4096, 256, 0, stream>>>((v4f*)d_out, n4, out, total);

    // 2) WMMA edge-block build + atomic scatter (128 edges per 256-thread block)
    const int blocks = (n_edges + 127) / 128;
    assemble_kernel<<<blocks, 256, 0, stream>>>(coord, delta, E, A, I, src, dst,
                                                out, n_edges, dim);
}